// Model_TrGNN_3161095930398
// MI455X (gfx1250) — compile-verified
//
#include <hip/hip_runtime.h>

// ---------------------------------------------------------------------------
// TrGNN forward for MI455X (gfx1250, wave32, WMMA).
//   N=2404, B=4, 75 demand hops, 3 status hops.
// - Demand hops: 75 sequential matvecs over L2-resident T (92.5MB < 192MB L2)
//   => bandwidth bound (2 FLOP / 4B). Split-K across 32 segments for
//   occupancy (2560 waves/hop), b128 coalesced column loads of T.
// - Dual-prop stages: real skinny GEMMs with shared W_norm =>
//   V_WMMA_F32_16X16X4_F32, branch-free inner loop (clamped addresses,
//   garbage rows/cols discarded at the predicated store).
// ---------------------------------------------------------------------------

#define NN     2404
#define BB     4
#define HOPS   75
#define KATT   76    // DEMAND_HOP + 1
#define C15    15
#define CIN    29
#define JSPLIT 32
#define SEG    76    // ceil(NN / JSPLIT); last segment = 48; both % 4 == 0

typedef __attribute__((ext_vector_type(2))) float v2f;
typedef __attribute__((ext_vector_type(8))) float v8f;

// --------------------------------------------------------------------------
// Setup: H[b][0][:] = X[b], and column 0 (the "y" column) of Xs1/Xs2/S.
// --------------------------------------------------------------------------
__global__ void k_setup(const float* __restrict__ X, float* __restrict__ H,
                        float* __restrict__ Xs1, float* __restrict__ Xs2,
                        float* __restrict__ S) {
    const int n = blockIdx.x * blockDim.x + threadIdx.x;
    const int b = blockIdx.y;
    if (n < NN) {
        const float x = X[(size_t)b * NN + n];
        H  [(size_t)b * KATT * NN + n] = x;
        Xs1[(size_t)b * 3    * NN + n] = x;
        Xs2[(size_t)b * 7    * NN + n] = x;
        S  [(size_t)b * C15  * NN + n] = x;
    }
}

// --------------------------------------------------------------------------
// Hop partial: part[s][b][n0..n0+3] = sum_{j in seg s} T[b][j][n] * y_{k-1}[j]
// (A = T_b^T => column access of T = contiguous across lanes, b128 loads.)
// --------------------------------------------------------------------------
__global__ void __launch_bounds__(128)
k_hop_part(const float* __restrict__ T, const float* __restrict__ H_all,
           float* __restrict__ part, int k) {
    __shared__ float ych[SEG];
    const int b  = blockIdx.z;
    const int s  = blockIdx.y;
    const int j0 = s * SEG;
    const int jm = ((NN - j0) < SEG) ? (NN - j0) : SEG;   // 76 or 48

    const float* __restrict__ yprev =
        H_all + (size_t)b * KATT * NN + (size_t)(k - 1) * NN;
    if (threadIdx.x < SEG) {
        const int j = j0 + threadIdx.x;
        ych[threadIdx.x] = (j < NN) ? yprev[j] : 0.f;
    }
    __syncthreads();

    const int n0 = 4 * (blockIdx.x * 128 + threadIdx.x);
    if (n0 >= NN) return;

    const float* __restrict__ tp = T + (size_t)b * NN * NN + (size_t)j0 * NN + n0;
    float ax = 0.f, ay = 0.f, az = 0.f, aw = 0.f;
    int jj = 0;
    for (; jj + 3 < jm; jj += 4) {
        const float4 t0 = *(const float4*)(tp + (size_t)(jj + 0) * NN);
        const float4 t1 = *(const float4*)(tp + (size_t)(jj + 1) * NN);
        const float4 t2 = *(const float4*)(tp + (size_t)(jj + 2) * NN);
        const float4 t3 = *(const float4*)(tp + (size_t)(jj + 3) * NN);
        const float y0 = ych[jj], y1 = ych[jj + 1], y2 = ych[jj + 2], y3 = ych[jj + 3];
        ax += t0.x * y0 + t1.x * y1 + t2.x * y2 + t3.x * y3;
        ay += t0.y * y0 + t1.y * y1 + t2.y * y2 + t3.y * y3;
        az += t0.z * y0 + t1.z * y1 + t2.z * y2 + t3.z * y3;
        aw += t0.w * y0 + t1.w * y1 + t2.w * y2 + t3.w * y3;
    }
    for (; jj < jm; jj++) {
        const float4 t = *(const float4*)(tp + (size_t)jj * NN);
        const float y = ych[jj];
        ax += t.x * y; ay += t.y * y; az += t.z * y; aw += t.w * y;
    }
    float* __restrict__ p = part + ((size_t)s * BB + b) * NN + n0;
    *(float4*)p = make_float4(ax, ay, az, aw);   // NN%4==0 -> 16B aligned
}

// --------------------------------------------------------------------------
// Hop reduce: y_k[n] = sum_s part[s][b][n]
// --------------------------------------------------------------------------
__global__ void __launch_bounds__(256)
k_hop_reduce(const float* __restrict__ part, float* __restrict__ H_all, int k) {
    const int n = blockIdx.x * 256 + threadIdx.x;
    const int b = blockIdx.y;
    if (n >= NN) return;
    float acc = 0.f;
#pragma unroll
    for (int s = 0; s < JSPLIT; s++)
        acc += part[((size_t)s * BB + b) * NN + n];
    H_all[(size_t)b * KATT * NN + (size_t)k * NN + n] = acc;
}

// --------------------------------------------------------------------------
// WMMA f32 16x16x4 skinny GEMM:  Out[b][c][n] = sum_j Aop[n][j] * Xin[b][c][j]
//   Aop = A (TRANSA=0) or A^T (TRANSA=1), A is NN x NN row-major.
// One wave per 16-row tile; single 16-col tile covers win<=7.
// Addresses are clamped (not predicated) so the K-loop is branch-free:
// garbage rows/cols only affect D elements never stored.
// --------------------------------------------------------------------------
template <bool TRANSA>
__global__ void __launch_bounds__(128)
k_gemm_wmma(const float* __restrict__ A, const float* __restrict__ Xin,
            float* __restrict__ Out, int win, int ldbB, int ldoB) {
    const int b    = blockIdx.y;
    const int wave = threadIdx.x >> 5;
    const int lane = threadIdx.x & 31;
    const int row0 = (blockIdx.x * 4 + wave) * 16;
    if (row0 >= NN) return;                 // wave-uniform exit

    const int half = lane >> 4;             // 0: K={0,1}, 1: K={2,3}
    const int m16  = lane & 15;

    int rowA = row0 + m16; if (rowA > NN - 1) rowA = NN - 1;   // clamp
    int cc   = m16;        if (cc   > win - 1) cc  = win - 1;  // clamp

    const float* __restrict__ Abase = TRANSA ? (A + rowA) : (A + (size_t)rowA * NN);
    const float* __restrict__ Xcol  = Xin + (size_t)b * ldbB + (size_t)cc * NN;

    v8f acc = {};
    for (int k0 = 0; k0 < NN; k0 += 4) {    // NN % 4 == 0
        const int kk = k0 + 2 * half;
        v2f a, bv;
        if (TRANSA) {                       // (A^T)[row][k] = A[k][row]
            a.x = Abase[(size_t)kk * NN];
            a.y = Abase[(size_t)(kk + 1) * NN];
        } else {
            const float2 t = *(const float2*)(Abase + kk);
            a.x = t.x; a.y = t.y;
        }
        const float2 u = *(const float2*)(Xcol + kk);
        bv.x = u.x; bv.y = u.y;
        acc = __builtin_amdgcn_wmma_f32_16x16x4_f32(
                  /*neg_a=*/false, a, /*neg_b=*/false, bv,
                  /*c_mod=*/(short)0, acc, /*reuse_a=*/false, /*reuse_b=*/false);
    }

    const bool cval = m16 < win;
    float* __restrict__ Ob = Out + (size_t)b * ldoB + (size_t)m16 * NN;
#pragma unroll
    for (int r = 0; r < 8; r++) {           // D: vgpr r -> M = r + 8*half
        const int nrow = row0 + r + 8 * half;
        if (cval && nrow < NN) Ob[nrow] = acc[r];
    }
}

// --------------------------------------------------------------------------
// att = softmax_o( S[b,n,:] @ att_w[n,:,:] + att_b[n,:] );  Hs = sum_o att*H.
// Two-pass streaming softmax (recompute logits) keeps register use low.
// --------------------------------------------------------------------------
__global__ void __launch_bounds__(256)
k_att(const float* __restrict__ S, const float* __restrict__ att_w,
      const float* __restrict__ att_b, const float* __restrict__ H,
      float* __restrict__ Hs) {
    const int n = blockIdx.x * blockDim.x + threadIdx.x;
    const int b = blockIdx.y;
    if (n >= NN) return;

    float s[C15];
#pragma unroll
    for (int ci = 0; ci < C15; ci++)
        s[ci] = S[(size_t)b * C15 * NN + (size_t)ci * NN + n];

    const float* __restrict__ aw = att_w + (size_t)n * C15 * KATT;
    const float* __restrict__ ab = att_b + (size_t)n * KATT;

    float mx = -3.402823466e38f;
    for (int o = 0; o < KATT; o++) {
        float l = ab[o];
#pragma unroll
        for (int ci = 0; ci < C15; ci++) l += s[ci] * aw[ci * KATT + o];
        mx = fmaxf(mx, l);
    }
    const float* __restrict__ Hb = H + (size_t)b * KATT * NN;
    float sum = 0.f, accv = 0.f;
    for (int o = 0; o < KATT; o++) {
        float l = ab[o];
#pragma unroll
        for (int ci = 0; ci < C15; ci++) l += s[ci] * aw[ci * KATT + o];
        const float e = __expf(l - mx);
        sum  += e;
        accv += e * Hb[(size_t)o * NN + n];
    }
    Hs[(size_t)b * NN + n] = accv / sum;
}

// --------------------------------------------------------------------------
// Y[n] = sum_c Hc[n][c]*out_w[n][c] + out_b[n]; Hc = [Hs.T | ToD | DoW]
// --------------------------------------------------------------------------
__global__ void __launch_bounds__(256)
k_out(const float* __restrict__ Hs, const float* __restrict__ ToD,
      const float* __restrict__ DoW, const float* __restrict__ out_w,
      const float* __restrict__ out_b, float* __restrict__ Y) {
    const int n = blockIdx.x * blockDim.x + threadIdx.x;
    if (n >= NN) return;
    const float* ow = out_w + (size_t)n * CIN;
    float acc = out_b[n];
#pragma unroll
    for (int b = 0; b < BB; b++) acc += Hs[(size_t)b * NN + n] * ow[b];
#pragma unroll
    for (int t = 0; t < 24; t++) acc += ToD[(size_t)n * 24 + t] * ow[4 + t];
    acc += DoW[n] * ow[28];
    Y[n] = acc;
}

// --------------------------------------------------------------------------
extern "C" void kernel_launch(void* const* d_in, const int* in_sizes, int n_in,
                              void* d_out, int out_size, void* d_ws, size_t ws_size,
                              hipStream_t stream) {
    const float* X    = (const float*)d_in[0];
    const float* T    = (const float*)d_in[1];
    // d_in[2] = W (unused by reference), d_in[3] = h_init (unused)
    const float* Wn   = (const float*)d_in[4];
    const float* ToD  = (const float*)d_in[5];
    const float* DoW  = (const float*)d_in[6];
    const float* attw = (const float*)d_in[7];
    const float* attb = (const float*)d_in[8];
    const float* outw = (const float*)d_in[9];
    const float* outb = (const float*)d_in[10];
    float* Y = (float*)d_out;

    float* ws    = (float*)d_ws;
    float* H     = ws;                               // B*76*N
    float* Xs1   = H     + (size_t)BB * KATT * NN;   // B*3*N
    float* Xs2   = Xs1   + (size_t)BB * 3 * NN;      // B*7*N
    float* S     = Xs2   + (size_t)BB * 7 * NN;      // B*15*N
    float* Hs    = S     + (size_t)BB * C15 * NN;    // B*N
    float* Hpart = Hs    + (size_t)BB * NN;          // JSPLIT*B*N

    const dim3 bs(256);
    const dim3 gsN((NN + 255) / 256, BB);
    k_setup<<<gsN, bs, 0, stream>>>(X, H, Xs1, Xs2, S);

    // 75 sequential hops: split-K partials + reduce (L2-resident T)
    const dim3 gPart((NN / 4 + 127) / 128, JSPLIT, BB);  // (5, 32, 4)
    const dim3 gRed((NN + 255) / 256, BB);
    for (int k = 1; k <= HOPS; k++) {
        k_hop_part  <<<gPart, 128, 0, stream>>>(T, H, Hpart, k);
        k_hop_reduce<<<gRed,  bs,  0, stream>>>(Hpart, H, k);
    }

    // dual-prop tree via WMMA skinny GEMMs
    const int  tiles = (NN + 15) / 16;               // 151
    const dim3 gG((tiles + 3) / 4, BB);
    k_gemm_wmma<false><<<gG, 128, 0, stream>>>(Wn, X,   Xs1 + 1 * NN, 1, NN,     3 * NN);
    k_gemm_wmma<true ><<<gG, 128, 0, stream>>>(Wn, X,   Xs1 + 2 * NN, 1, NN,     3 * NN);
    k_gemm_wmma<false><<<gG, 128, 0, stream>>>(Wn, Xs1, Xs2 + 1 * NN, 3, 3 * NN, 7 * NN);
    k_gemm_wmma<true ><<<gG, 128, 0, stream>>>(Wn, Xs1, Xs2 + 4 * NN, 3, 3 * NN, 7 * NN);
    k_gemm_wmma<false><<<gG, 128, 0, stream>>>(Wn, Xs2, S   + 1 * NN, 7, 7 * NN, 15 * NN);
    k_gemm_wmma<true ><<<gG, 128, 0, stream>>>(Wn, Xs2, S   + 8 * NN, 7, 7 * NN, 15 * NN);

    k_att<<<gsN, bs, 0, stream>>>(S, attw, attb, H, Hs);

    const dim3 g1((NN + 255) / 256);
    k_out<<<g1, bs, 0, stream>>>(Hs, ToD, DoW, outw, outb, Y);
}